// BERTNED_48387101557005
// MI455X (gfx1250) — compile-verified
//
#include <hip/hip_runtime.h>

typedef float v2f __attribute__((ext_vector_type(2)));
typedef float v4f __attribute__((ext_vector_type(4)));
typedef float v8f __attribute__((ext_vector_type(8)));

#define B_  64
#define N_  512
#define L_  768
#define M_  24
#define K_  30
#define H_  1024
#define R_  (B_ * M_)     // 1536 alias rows
#define KD_ (2 * L_)      // 1536 GEMM reduction dim

// ---------------------------------------------------------------------------
// Kernel 1: gather start/end word vectors into pair matrix (R_ x KD_)
// ---------------------------------------------------------------------------
__global__ __launch_bounds__(192)
void gather_pair_kernel(const float* __restrict__ sent,   // (B,N,L)
                        const int* __restrict__ sidx,     // (B,M)
                        const int* __restrict__ eidx,     // (B,M)
                        float* __restrict__ pair) {       // (R_, KD_)
  int r = blockIdx.x;            // 0..R_-1  (b*M + m)
  int b = r / M_;
  int t = threadIdx.x;           // 0..191 -> one float4 of the 768-wide half
  int s = sidx[r];
  int e = eidx[r];
  const float4* srow = (const float4*)(sent + ((size_t)b * N_ + (s < 0 ? 0 : s)) * L_);
  const float4* erow = (const float4*)(sent + ((size_t)b * N_ + (e < 0 ? 0 : e)) * L_);
  float4 z = make_float4(0.f, 0.f, 0.f, 0.f);
  float4 sv = (s < 0) ? z : srow[t];
  float4 ev = (e < 0) ? z : erow[t];
  float4* prow = (float4*)(pair + (size_t)r * KD_);
  prow[t] = sv;             // cols [0, 768)
  prow[t + L_ / 4] = ev;    // cols [768, 1536)
}

// ---------------------------------------------------------------------------
// Kernel 2: alias_emb = pair @ W_span + b_span   via V_WMMA_F32_16X16X4_F32
//   block = 256 threads (8 waves), tile 128x128, BK = 32
// ---------------------------------------------------------------------------
#define BM 128
#define BN 128
#define BK 32
#define LDA 36     // 32 + 4 pad: conflict-free + float4-aligned
#define LDB 36

__global__ __launch_bounds__(256)
void span_gemm_kernel(const float* __restrict__ pair,   // (R_, KD_)
                      const float* __restrict__ W,      // (KD_, H_)
                      const float* __restrict__ bias,   // (H_)
                      float* __restrict__ alias_out) {  // (R_, H_)
  __shared__ float As[BM * LDA];   // 18 KB
  __shared__ float Bt[BN * LDB];   // 18 KB (B stored transposed: Bt[n][k])

  const int tid  = threadIdx.x;
  const int wid  = tid >> 5;       // wave 0..7 -> 16-row strip
  const int lane = tid & 31;
  const int lr    = lane & 15;          // row/col within 16
  const int khalf = (lane >> 4) << 1;   // lanes 16..31 hold K+2,K+3

  const int rowBase = blockIdx.y * BM;
  const int colBase = blockIdx.x * BN;

  v8f acc[8];
  const v8f vzero = {0.f, 0.f, 0.f, 0.f, 0.f, 0.f, 0.f, 0.f};
#pragma unroll
  for (int i = 0; i < 8; ++i) acc[i] = vzero;

  for (int kc = 0; kc < KD_; kc += BK) {
    // ---- stage A: 128x32 (1024 float4, 4 per thread), row-major ----
#pragma unroll
    for (int i = 0; i < 4; ++i) {
      int idx  = tid + i * 256;        // float4 index in chunk
      int arow = idx >> 3;             // 8 float4 per row
      int acv  = idx & 7;
      float4 v = *(const float4*)(pair + (size_t)(rowBase + arow) * KD_ + kc + acv * 4);
      *(float4*)(&As[arow * LDA + acv * 4]) = v;
    }
    // ---- stage B transposed: W[kc+k][colBase+n] -> Bt[n][k] ----
#pragma unroll
    for (int i = 0; i < 4; ++i) {
      int fidx = tid + i * 256;        // float4 index in chunk
      int brow = fidx >> 5;            // k  (32 float4 per W row-chunk)
      int bnv  = fidx & 31;            // n / 4
      float4 v = *(const float4*)(W + (size_t)(kc + brow) * H_ + colBase + bnv * 4);
      Bt[(bnv * 4 + 0) * LDB + brow] = v.x;
      Bt[(bnv * 4 + 1) * LDB + brow] = v.y;
      Bt[(bnv * 4 + 2) * LDB + brow] = v.z;
      Bt[(bnv * 4 + 3) * LDB + brow] = v.w;
    }
    __syncthreads();

    // ---- 8 k-steps; per step: batch-load A + 8 B frags, then 8 WMMAs ----
#pragma unroll
    for (int kk = 0; kk < BK; kk += 4) {
      v2f a = *(const v2f*)(&As[(wid * 16 + lr) * LDA + kk + khalf]);
      v2f bf[8];
#pragma unroll
      for (int nt = 0; nt < 8; ++nt)
        bf[nt] = *(const v2f*)(&Bt[(nt * 16 + lr) * LDB + kk + khalf]);
#pragma unroll
      for (int nt = 0; nt < 8; ++nt)
        acc[nt] = __builtin_amdgcn_wmma_f32_16x16x4_f32(
            false, a, false, bf[nt], (short)0, acc[nt], false, false);
    }
    __syncthreads();
  }

  // ---- epilogue: bias add + store (C/D layout: VGPR v -> rows v / v+8) ----
  const int rhi = (lane >> 4) << 3;    // +8 rows for lanes 16..31
#pragma unroll
  for (int nt = 0; nt < 8; ++nt) {
    int col = colBase + nt * 16 + lr;
    float bv = bias[col];
#pragma unroll
    for (int v = 0; v < 8; ++v) {
      int row = rowBase + wid * 16 + v + rhi;
      alias_out[(size_t)row * H_ + col] = acc[nt][v] + bv;
    }
  }
}

// ---------------------------------------------------------------------------
// Kernel 3 (bandwidth-bound, ~378 MB => the runtime): streams entity rows
// once with NON-TEMPORAL load/store (no reuse; don't thrash 192MB L2),
// fuses ent = mask ? 0 : entity  and  scores = dot(alias, ent_row).
// ---------------------------------------------------------------------------
__global__ __launch_bounds__(256)
void mask_score_kernel(const float* __restrict__ ent_in,        // (B,M,K,H)
                       const unsigned char* __restrict__ mask,  // (B,M,K) bool
                       const float* __restrict__ alias_emb,     // (R_, H_)
                       float* __restrict__ out_scores,          // (B,M,K)
                       float* __restrict__ out_ent) {           // (B,M,K,H)
  const int r    = blockIdx.x;       // b*M + m
  const int t    = threadIdx.x;      // covers float4 t of 256
  const int wid  = t >> 5;
  const int lane = t & 31;
  __shared__ float red[8 * K_];

  // alias row was just produced by the GEMM -> likely L2-resident: regular load
  v4f av = ((const v4f*)(alias_emb + (size_t)r * H_))[t];

  float part[K_];
#pragma unroll
  for (int k = 0; k < K_; ++k) {
    size_t base = ((size_t)r * K_ + k) * H_;
    v4f e = __builtin_nontemporal_load(((const v4f*)(ent_in + base)) + t);
    if (mask[r * K_ + k]) e = (v4f){0.f, 0.f, 0.f, 0.f};
    __builtin_nontemporal_store(e, ((v4f*)(out_ent + base)) + t);
    part[k] = av.x * e.x + av.y * e.y + av.z * e.z + av.w * e.w;
  }

#pragma unroll
  for (int k = 0; k < K_; ++k) {
    float p = part[k];
#pragma unroll
    for (int off = 16; off > 0; off >>= 1) p += __shfl_xor(p, off, 32);
    if (lane == 0) red[wid * K_ + k] = p;
  }
  __syncthreads();
  if (t < K_) {
    float s = 0.f;
#pragma unroll
    for (int w = 0; w < 8; ++w) s += red[w * K_ + t];
    out_scores[r * K_ + t] = s;
  }
}

// ---------------------------------------------------------------------------
extern "C" void kernel_launch(void* const* d_in, const int* in_sizes, int n_in,
                              void* d_out, int out_size, void* d_ws, size_t ws_size,
                              hipStream_t stream) {
  const float*         sent = (const float*)d_in[0];         // (B,N,L)
  const float*         ent  = (const float*)d_in[1];         // (B,M,K,H)
  const unsigned char* msk  = (const unsigned char*)d_in[2]; // (B,M,K) bool
  const int*           sidx = (const int*)d_in[3];           // (B,M)
  const int*           eidx = (const int*)d_in[4];           // (B,M)
  const float*         Wsp  = (const float*)d_in[5];         // (2L,H)
  const float*         bsp  = (const float*)d_in[6];         // (H)

  float* out        = (float*)d_out;
  float* out_scores = out;                         // (B,M,K) = 46080
  float* out_ent    = out + (size_t)R_ * K_;       // (B,M,K,H)

  float* pairW  = (float*)d_ws;                    // 1536*1536 f32
  float* aliasW = pairW + (size_t)R_ * KD_;        // 1536*1024 f32

  gather_pair_kernel<<<R_, 192, 0, stream>>>(sent, sidx, eidx, pairW);

  dim3 g2(H_ / BN, R_ / BM);                       // 8 x 12 blocks
  span_gemm_kernel<<<g2, 256, 0, stream>>>(pairW, Wsp, bsp, aliasW);

  mask_score_kernel<<<R_, 256, 0, stream>>>(ent, msk, aliasW, out_scores, out_ent);
}